// MarkovChain_50620484551201
// MI455X (gfx1250) — compile-verified
//
#include <hip/hip_runtime.h>
#include <hip/hip_bf16.h>
#include <stdint.h>

#define B_ 64
#define T_ 1024
#define K_ 512
#define EPSF 1e-8f
#define LOG_EPSF 1e-20f
#define PITCH 516   // 512 + 4 floats pad: kills LDS bank conflicts for A and B^T reads

typedef __attribute__((ext_vector_type(2))) float v2f;
typedef __attribute__((ext_vector_type(8))) float v8f;
typedef __attribute__((ext_vector_type(4))) unsigned int u32x4;
typedef __attribute__((ext_vector_type(4))) int i32x4;
typedef __attribute__((ext_vector_type(8))) int i32x8;

// ---------------------------------------------------------------------------
// threefry2x32, bit-compatible with JAX (20 rounds, rotations 13,15,26,6 /
// 17,29,16,24, key schedule with 0x1BD11BDA parity word).
// ---------------------------------------------------------------------------
__device__ __forceinline__ uint32_t rotl32(uint32_t v, int n) {
  return (v << n) | (v >> (32 - n));
}

__device__ void threefry2x32(uint32_t k0, uint32_t k1, uint32_t x0, uint32_t x1,
                             uint32_t& o0, uint32_t& o1) {
  uint32_t ks2 = 0x1BD11BDAu ^ k0 ^ k1;
  x0 += k0; x1 += k1;
#define TF_R4(a, b, c, d)                     \
  x0 += x1; x1 = rotl32(x1, a); x1 ^= x0;     \
  x0 += x1; x1 = rotl32(x1, b); x1 ^= x0;     \
  x0 += x1; x1 = rotl32(x1, c); x1 ^= x0;     \
  x0 += x1; x1 = rotl32(x1, d); x1 ^= x0;
  TF_R4(13, 15, 26, 6);  x0 += k1;  x1 += ks2 + 1u;
  TF_R4(17, 29, 16, 24); x0 += ks2; x1 += k0 + 2u;
  TF_R4(13, 15, 26, 6);  x0 += k0;  x1 += k1 + 3u;
  TF_R4(17, 29, 16, 24); x0 += k1;  x1 += ks2 + 4u;
  TF_R4(13, 15, 26, 6);  x0 += ks2; x1 += k0 + 5u;
#undef TF_R4
  o0 = x0; o1 = x1;
}

// random_bits(key, 32, (n,)) element m, for even n: pair = (m, m+n/2) if m<n/2
// (take first output), else pair = (m-n/2, m) (take second output).
__device__ __forceinline__ uint32_t jax_bits(uint32_t kh, uint32_t kl,
                                             uint32_t m, uint32_t half) {
  uint32_t x0, x1, o0, o1;
  int sel;
  if (m < half) { x0 = m; x1 = m + half; sel = 0; }
  else          { x0 = m - half; x1 = m; sel = 1; }
  threefry2x32(kh, kl, x0, x1, o0, o1);
  return sel ? o1 : o0;
}

// jax.random.gumbel: -log(-log(uniform(key, minval=tiny, maxval=1)))
__device__ __forceinline__ float gumbel_from_bits(uint32_t bits) {
  const float tinyf = 1.17549435e-38f;
  float u01 = __uint_as_float((bits >> 9) | 0x3f800000u) - 1.0f;
  float u = fmaxf(tinyf, u01 * (1.0f - tinyf) + tinyf);
  return -logf(-logf(u));
}

// ---------------------------------------------------------------------------
// msg0 = init_p*(1-m) + m*onehot, normalized with +EPS. Also resets the
// forward-barrier counter (deterministic across graph replays).
// ---------------------------------------------------------------------------
__global__ void __launch_bounds__(256)
hmm_init_kernel(const int* __restrict__ data, const float* __restrict__ masks,
                const float* __restrict__ initp, float* __restrict__ msgs,
                unsigned int* __restrict__ syncv) {
  const int b = blockIdx.x;
  const int tid = threadIdx.x;
  __shared__ float red[256];
  float part = 0.f;
  for (int k = tid; k < K_; k += 256) part += initp[k];
  red[tid] = part; __syncthreads();
  for (int s = 128; s > 0; s >>= 1) {
    if (tid < s) red[tid] += red[tid + s];
    __syncthreads();
  }
  const float sinit = red[0];
  const float m0 = masks[b * T_ + 0];
  const int d0 = data[b * T_ + 0];
  for (int k = tid; k < K_; k += 256) {
    float v;
    if (m0 > 0.f) v = (k == d0 ? 1.0f : 0.0f) / (1.0f + EPSF);
    else          v = initp[k] / (sinit + EPSF);
    msgs[(size_t)b * K_ + k] = v;
  }
  if (b == 0 && tid == 0) *syncv = 0u;
}

// ---------------------------------------------------------------------------
// Forward filter: persistent kernel, grid = 4 N-chunks x 4 M-tiles = 16 WGs.
// Each WG keeps P^T[n0..n0+127][0..511] resident in LDS for the whole scan.
// Per step the 16x512 prev-message tile is DMA'd into LDS by the Tensor Data
// Mover (TENSOR_LOAD_TO_LDS, TENSORcnt), with D# padding (2 dwords per 256)
// reproducing the 516-float bank-conflict-free pitch. The 16x128 slice of
// msg_t is computed with V_WMMA_F32_16X16X4_F32. Device-wide barrier per step.
// ---------------------------------------------------------------------------
__global__ void __launch_bounds__(256)
hmm_forward_kernel(const int* __restrict__ data, const float* __restrict__ masks,
                   const float* __restrict__ Pg, float* __restrict__ msgs,
                   unsigned int* __restrict__ syncv) {
  extern __shared__ float smem[];
  float* PbufT  = smem;                    // [128][PITCH]  P^T chunk (no intra-row pad)
  float* Abuf   = smem + 128 * PITCH;      // TDM tile: offs = r*516 + c + 2*(c>=256)
  float* sums   = Abuf + 16 * PITCH;       // [16]
  float* rscale = sums + 16;               // [16]
  float* maskr  = rscale + 16;             // [16]
  int*   datar  = (int*)(maskr + 16);      // [16]

  const int tid  = threadIdx.x;
  const int lane = tid & 31;
  const int wv   = tid >> 5;               // 8 waves -> 8 N-subtiles of 16
  const int n0   = blockIdx.x * 128;       // column chunk
  const int r0   = blockIdx.y * 16;        // row (batch) tile

  // Stage transposed P chunk into LDS once (resident for all T steps).
  for (int idx = tid; idx < K_ * 128; idx += 256) {
    const int k = idx >> 7, c = idx & 127;
    PbufT[c * PITCH + k] = Pg[k * K_ + n0 + c];
  }

  // WMMA fragment coordinates (fp32 16x16x4 layout, ISA 7.12.2):
  //  A lane L: row = L&15, holds K = k0 + 2*(L>>4) + {0,1}
  //  B lane L: col = L&15, holds K = k0 + 2*(L>>4) + {0,1}
  //  D vgpr j, lane L: M = j + 8*(L>>4), N = L&15
  const int halfsel = lane >> 4;
  const int Ncol    = lane & 15;
  const int koff    = halfsel * 2;

  // Tensor DMA descriptor pieces that never change (ISA §8.3/8.4):
  // group1: data_size=4B(2), pad_enable, pad_interval=256dw(7), pad_amount=2dw(1)
  //         tensor_dim0=512, tensor_dim1=16, tile_dim0=512, tile_dim1=16,
  //         tensor_dim0_stride=512.
  const i32x8 tdm_g1 = { (int)0x03D20000, (int)(512 << 16), (int)(16 << 16),
                         (int)(512 << 16), 16, 512, 0, 0 };
  const i32x4 tdm_z4 = { 0, 0, 0, 0 };
#if defined(__clang_major__) && __clang_major__ >= 23
  const i32x8 tdm_z8 = { 0, 0, 0, 0, 0, 0, 0, 0 };
#endif
  const uint32_t lds_abuf = (uint32_t)(uintptr_t)Abuf;   // LDS byte address

  __syncthreads();

  for (int t = 1; t < T_; ++t) {
    if (tid < 16) {
      sums[tid]  = 0.f;
      maskr[tid] = masks[(r0 + tid) * T_ + t];
      datar[tid] = data[(r0 + tid) * T_ + t];
    }
    // --- TDM: DMA prev-message tile (16 x 512 f32, row stride 512) into LDS.
    if (tid < 32) {                         // one wave owns the tensor op
      const uint64_t gaddr =
          (uint64_t)(uintptr_t)(msgs + (size_t)(t - 1) * (B_ * K_) + (size_t)r0 * K_);
      u32x4 g0;
      g0.x = 1u;                                        // count=1 valid descriptor
      g0.y = lds_abuf;                                  // lds_addr
      g0.z = (uint32_t)gaddr;                           // global_addr[31:0]
      g0.w = (uint32_t)(gaddr >> 32) | (2u << 30);      // global_addr[56:32] | type=2
#if defined(__clang_major__) && __clang_major__ >= 23
      __builtin_amdgcn_tensor_load_to_lds(g0, tdm_g1, tdm_z4, tdm_z4, tdm_z8, 0);
#else
      __builtin_amdgcn_tensor_load_to_lds(g0, tdm_g1, tdm_z4, tdm_z4, 0);
#endif
      __builtin_amdgcn_s_wait_tensorcnt(0);
    }
    __syncthreads();                        // publish LDS tile to all 8 waves

    // Exact row sums of prev tile (for on-the-fly renormalization).
    {
      const int r = tid & 15, seg = tid >> 4;
      const float* ap = Abuf + r * PITCH + seg * 32 + (seg >= 8 ? 2 : 0);
      float p = 0.f;
#pragma unroll
      for (int j = 0; j < 32; ++j) p += ap[j];
      atomicAdd(&sums[r], p);
    }
    __syncthreads();
    if (tid < 16) rscale[tid] = 1.0f / sums[tid];
    __syncthreads();

    // 16x16 tile = sum over K of A(16x4) x B(4x16), fp32 WMMA.
    v8f acc = {};
    const float* arow_p = Abuf + (lane & 15) * PITCH + koff;
    const float* bcol_p = PbufT + (wv * 16 + Ncol) * PITCH + koff;
#pragma unroll 4
    for (int k0 = 0; k0 < 256; k0 += 4) {
      v2f a = *(const v2f*)(arow_p + k0);
      v2f b = *(const v2f*)(bcol_p + k0);
      acc = __builtin_amdgcn_wmma_f32_16x16x4_f32(
          false, a, false, b, (short)0, acc, false, false);
    }
#pragma unroll 4
    for (int k0 = 256; k0 < 512; k0 += 4) {
      v2f a = *(const v2f*)(arow_p + k0 + 2);   // +2: TDM pad bump at column 256
      v2f b = *(const v2f*)(bcol_p + k0);
      acc = __builtin_amdgcn_wmma_f32_16x16x4_f32(
          false, a, false, b, (short)0, acc, false, false);
    }

    // Branchless: scale by 1/rowsum, overwrite masked rows with one-hot, store.
    float rs8[8], mk8[8];
    int dt8[8];
#pragma unroll
    for (int j = 0; j < 8; ++j) {
      const int M = j + halfsel * 8;
      rs8[j] = rscale[M]; mk8[j] = maskr[M]; dt8[j] = datar[M];
    }
    float* out = msgs + (size_t)t * (B_ * K_);
    const int cg = n0 + wv * 16 + Ncol;
#pragma unroll
    for (int j = 0; j < 8; ++j) {
      const int M = j + halfsel * 8;
      const float scaled = acc[j] * rs8[j];
      const float onehot = (dt8[j] == cg) ? 1.0f : 0.0f;
      const float v = (mk8[j] > 0.f) ? onehot : scaled;
      out[(size_t)(r0 + M) * K_ + cg] = v;
    }

    // Device-scope barrier: all 16 WGs finished step t.
    __syncthreads();
    if (tid == 0) {
      asm volatile("s_wait_storecnt 0x0" ::: "memory");
      __hip_atomic_fetch_add(syncv, 1u, __ATOMIC_RELEASE, __HIP_MEMORY_SCOPE_AGENT);
      const unsigned int target = 16u * (unsigned int)t;
      while (__hip_atomic_load(syncv, __ATOMIC_ACQUIRE, __HIP_MEMORY_SCOPE_AGENT) < target)
        __builtin_amdgcn_s_sleep(2);
    }
    __syncthreads();
  }
}

// ---------------------------------------------------------------------------
// Exact renormalization of msg_t (t>=1) to match reference-stored messages.
// ---------------------------------------------------------------------------
__global__ void __launch_bounds__(256)
hmm_norm_kernel(float* __restrict__ msgs) {
  const int row = blockIdx.x;                 // 0..(T-1)*B-1
  const int t = 1 + row / B_, b = row % B_;
  float* m = msgs + ((size_t)t * B_ + b) * K_;
  __shared__ float red[256];
  const int tid = threadIdx.x;
  red[tid] = m[tid] + m[tid + 256];
  __syncthreads();
  for (int s = 128; s > 0; s >>= 1) {
    if (tid < s) red[tid] += red[tid + s];
    __syncthreads();
  }
  const float S = red[0];
  m[tid] /= S;
  m[tid + 256] /= S;
}

// ---------------------------------------------------------------------------
// Backward sampling. One block per chain (b,i); K=512 handled by 256 threads.
// RNG replicates: key(42) -> split -> (k_last, k_seq); split(k_seq, T-1);
// categorical = argmax(log(w)+gumbel), first-index tie break.
// ---------------------------------------------------------------------------
__global__ void __launch_bounds__(256)
hmm_backward_kernel(const int* __restrict__ data, const float* __restrict__ masks,
                    const float* __restrict__ Pg, const float* __restrict__ msgs,
                    float* __restrict__ outp, int I) {
  const int chain = blockIdx.x;   // b*I + i
  const int b = chain / I;
  const int tid = threadIdx.x;
  const uint32_t halfN = (uint32_t)(B_ * I * K_) / 2u;

  __shared__ float rv[256];
  __shared__ int   ri[256];
  __shared__ float rs[256];
  __shared__ uint32_t keybuf[2];

  // key(42) = (0,42); split -> k_last=(f0(0,2), f0(1,3)), k_seq=(f1(0,2), f1(1,3))
  uint32_t a0, a1, b0, b1;
  threefry2x32(0u, 42u, 0u, 2u, a0, a1);
  threefry2x32(0u, 42u, 1u, 3u, b0, b1);
  const uint32_t kl_hi = a0, kl_lo = b0;   // k_last
  const uint32_t ks_hi = a1, ks_lo = b1;   // k_seq

  // ---- last timestep: categorical(log(msg_{T-1}+1e-20)) per (b,i)
  const float* mlast = msgs + ((size_t)(T_ - 1) * B_ + b) * K_;
  float bestv = -3.4e38f; int besti = 0;
#pragma unroll
  for (int kk = 0; kk < 2; ++kk) {
    const int k = tid + kk * 256;
    const uint32_t m = (uint32_t)chain * (uint32_t)K_ + (uint32_t)k;
    const float v = logf(mlast[k] + LOG_EPSF) + gumbel_from_bits(jax_bits(kl_hi, kl_lo, m, halfN));
    if (v > bestv || (v == bestv && k < besti)) { bestv = v; besti = k; }
  }
  rv[tid] = bestv; ri[tid] = besti; __syncthreads();
  for (int s = 128; s > 0; s >>= 1) {
    if (tid < s) {
      const float vo = rv[tid + s]; const int io = ri[tid + s];
      if (vo > rv[tid] || (vo == rv[tid] && io < ri[tid])) { rv[tid] = vo; ri[tid] = io; }
    }
    __syncthreads();
  }
  int s_next;
  {
    const int s_samp = ri[0];
    const float mL = masks[b * T_ + (T_ - 1)];
    const int dL = data[b * T_ + (T_ - 1)];
    s_next = (mL > 0.f) ? s_samp : dL;   // as written in the reference
  }
  if (tid == 0) outp[(size_t)chain * T_ + (T_ - 1)] = (float)s_next;

  const float invK = 1.0f / (float)K_;
  for (int t = T_ - 2; t >= 0; --t) {
    const int j = (T_ - 2) - t;          // scan-step key index
    if (tid == 0) {
      // key_j = (bits[2j], bits[2j+1]) of split(k_seq, T-1), n = 2*(T-1) counters
      const uint32_t H = (uint32_t)(T_ - 1);
      keybuf[0] = jax_bits(ks_hi, ks_lo, 2u * (uint32_t)j, H);
      keybuf[1] = jax_bits(ks_hi, ks_lo, 2u * (uint32_t)j + 1u, H);
    }
    __syncthreads();
    const uint32_t kh = keybuf[0], klo = keybuf[1];

    const float* msg = msgs + ((size_t)t * B_ + b) * K_;
    float w[2];
    float psum = 0.f;
#pragma unroll
    for (int kk = 0; kk < 2; ++kk) {
      const int k = tid + kk * 256;
      const float trans = Pg[(size_t)k * K_ + s_next];   // P^T[s_next][k] = P[k][s_next]
      w[kk] = msg[k] * (trans + invK);
      psum += w[kk];
    }
    rs[tid] = psum; __syncthreads();
    for (int s = 128; s > 0; s >>= 1) {
      if (tid < s) rs[tid] += rs[tid + s];
      __syncthreads();
    }
    const float S = rs[0] + EPSF;

    bestv = -3.4e38f; besti = 0;
#pragma unroll
    for (int kk = 0; kk < 2; ++kk) {
      const int k = tid + kk * 256;
      const uint32_t m = (uint32_t)chain * (uint32_t)K_ + (uint32_t)k;
      const float v = logf(w[kk] / S + LOG_EPSF) +
                      gumbel_from_bits(jax_bits(kh, klo, m, halfN));
      if (v > bestv || (v == bestv && k < besti)) { bestv = v; besti = k; }
    }
    __syncthreads();
    rv[tid] = bestv; ri[tid] = besti; __syncthreads();
    for (int s = 128; s > 0; s >>= 1) {
      if (tid < s) {
        const float vo = rv[tid + s]; const int io = ri[tid + s];
        if (vo > rv[tid] || (vo == rv[tid] && io < ri[tid])) { rv[tid] = vo; ri[tid] = io; }
      }
      __syncthreads();
    }
    const int smp = ri[0];
    const float mt = masks[b * T_ + t];
    const int dt = data[b * T_ + t];
    s_next = (mt > 0.f) ? dt : smp;      // bstep: masked -> observed data
    if (tid == 0) outp[(size_t)chain * T_ + t] = (float)s_next;
    __syncthreads();
  }
}

// ---------------------------------------------------------------------------
extern "C" void kernel_launch(void* const* d_in, const int* in_sizes, int n_in,
                              void* d_out, int out_size, void* d_ws, size_t ws_size,
                              hipStream_t stream) {
  const int*   data  = (const int*)d_in[0];
  const float* masks = (const float*)d_in[1];
  const float* initp = (const float*)d_in[2];
  const float* Pg    = (const float*)d_in[3];
  const int I = out_size / (B_ * T_);          // nb_imputation (4)

  unsigned char* ws = (unsigned char*)d_ws;
  unsigned int* syncv = (unsigned int*)ws;     // barrier counter
  float* msgs = (float*)(ws + 256);            // [T][B][K] messages, 128 MB

  hmm_init_kernel<<<dim3(B_), dim3(256), 0, stream>>>(data, masks, initp, msgs, syncv);

  const size_t shmem = (size_t)(128 * PITCH + 16 * PITCH + 64) * sizeof(float); // ~297 KB
  hmm_forward_kernel<<<dim3(4, 4), dim3(256), shmem, stream>>>(data, masks, Pg, msgs, syncv);

  hmm_norm_kernel<<<dim3((T_ - 1) * B_), dim3(256), 0, stream>>>(msgs);

  hmm_backward_kernel<<<dim3(B_ * I), dim3(256), 0, stream>>>(data, masks, Pg, msgs,
                                                              (float*)d_out, I);
}